// OptimizedS4Layer_10282151707170
// MI455X (gfx1250) — compile-verified
//
#include <hip/hip_runtime.h>
#include <math.h>

// Problem constants (reference: H=512, L=16384, N=64)
#define HH 512
#define LL 16384
#define NN 64
#define L2 32768   // 2*L

typedef __attribute__((ext_vector_type(2))) float v2f;
typedef __attribute__((ext_vector_type(8))) float v8f;
typedef __attribute__((ext_vector_type(4))) unsigned int v4u;
typedef __attribute__((ext_vector_type(4))) int v4i;
typedef __attribute__((ext_vector_type(8))) int v8i;

#if defined(__has_builtin)
#if __has_builtin(__builtin_amdgcn_wmma_f32_16x16x4_f32)
#define USE_WMMA_F32 1
#endif
#if __has_builtin(__builtin_amdgcn_tensor_load_to_lds)
#define HAVE_TDM 1
#endif
#endif

__device__ __forceinline__ float2 cmulf(float2 a, float2 b) {
  return make_float2(a.x * b.x - a.y * b.y, a.x * b.y + a.y * b.x);
}
__device__ __forceinline__ float2 caddf(float2 a, float2 b) { return make_float2(a.x + b.x, a.y + b.y); }
__device__ __forceinline__ float2 csubf(float2 a, float2 b) { return make_float2(a.x - b.x, a.y - b.y); }

// ---------------------------------------------------------------------------
// Kernel 1: Ct = C @ M^T ; A0 = Ct*B ; A1 = Ct*p ; A2 = q*B ; qp = q*p
// ---------------------------------------------------------------------------
__global__ __launch_bounds__(256) void s4_prep(
    const float* __restrict__ Bm, const float* __restrict__ Cm,
    const float* __restrict__ Mm, const float* __restrict__ p,
    const float* __restrict__ q, float* __restrict__ A0,
    float* __restrict__ A1, float* __restrict__ A2, float* __restrict__ qp) {
  int idx = blockIdx.x * 256 + threadIdx.x;  // H*N threads
  int h = idx >> 6, n = idx & 63;
  const float* Mrow = Mm + n * NN;
  const float* Crow = Cm + h * NN;
  float ct = 0.f;
#pragma unroll 8
  for (int m = 0; m < NN; ++m) ct = fmaf(Mrow[m], Crow[m], ct);
  float b = Bm[h * NN + n];
  A0[idx] = ct * b;
  A1[idx] = ct * p[n];
  A2[idx] = q[n] * b;
  if (h == 0) qp[n] = q[n] * p[n];
}

// ---------------------------------------------------------------------------
// Kernel 2: cauchy re/im (L x N), k11[l] = sum_n qp[n]*cauchy[l,n], w2[l]=2/(1+r)
// f_l = 2(1-r)/(1+r) = i * (-2 sin th / (1 + cos th)),  r = exp(i th)
// ---------------------------------------------------------------------------
__global__ __launch_bounds__(64) void s4_cauchy_init(
    const float* __restrict__ G /* complex64 interleaved, N */,
    const float* __restrict__ qp, float* __restrict__ cr, float* __restrict__ ci,
    float* __restrict__ k11, float* __restrict__ w2) {
  int l = blockIdx.x;
  int n = threadIdx.x;
  float th = 6.28318530717958647692f * (float)l / (float)LL;
  float c = cosf(th), s = sinf(th);
  float den1 = fmaxf(1.f + c, 1e-20f);
  float fi = -2.f * s / den1;       // f_l = i*fi
  float gr = G[2 * n], gi = G[2 * n + 1];
  float x = -gr, yy = fi - gi;      // f - Gamma = x + i*yy
  float d = 1.f / (x * x + yy * yy);
  float crv = x * d, civ = -yy * d; // 1/(x+iy)
  cr[l * NN + n] = crv;
  ci[l * NN + n] = civ;
  __shared__ float sr[64], si[64];
  float qn = qp[n];
  sr[n] = qn * crv;
  si[n] = qn * civ;
  __syncthreads();
  for (int off = 32; off > 0; off >>= 1) {
    if (n < off) { sr[n] += sr[n + off]; si[n] += si[n + off]; }
    __syncthreads();
  }
  if (n == 0) {
    k11[2 * l] = sr[0];
    k11[2 * l + 1] = si[0];
    w2[2 * l] = 1.f;                 // 2/(1+r) = 1 - i*tan(th/2)
    w2[2 * l + 1] = -s / den1;
  }
}

// ---------------------------------------------------------------------------
// Kernel 3: WMMA f32 GEMM over K=64.
// All 8 waves of a block share the same 16-row h-tile of A0/A1/A2, so the
// three 16x64 A tiles are staged once into LDS by a single 3-D TDM descriptor
// (tile 64x16x3, z-stride = H*N between the A0/A1/A2 planes). B tiles
// (cauchy re/im) stream straight from global (L2-resident, pipelined vs WMMA).
// ---------------------------------------------------------------------------
__global__ __launch_bounds__(256) void s4_cauchy_gemm(
    const float* __restrict__ A0, const float* __restrict__ A1,
    const float* __restrict__ A2, const float* __restrict__ cr,
    const float* __restrict__ ci, const float* __restrict__ k11,
    const float* __restrict__ w2, float2* __restrict__ outAt) {
  int lane = threadIdx.x & 31;
  int wave = threadIdx.x >> 5;
  int gw = blockIdx.x * 8 + wave;     // 32768 tiles total
  int tl = gw & 1023;                 // 1024 l-tiles
  int th = gw >> 10;                  // 32 h-tiles (uniform across the block)
  int h0 = th << 4, l0 = tl << 4;
  int hf = lane >> 4;                 // half-wave select (K pairs / M+8 rows)
  int lr = lane & 15;

  const float2* pbr = (const float2*)(cr + (l0 + lr) * NN + 2 * hf);
  const float2* pbi = (const float2*)(ci + (l0 + lr) * NN + 2 * hf);

  v8f c00r = {}, c00i = {}, c01r = {}, c01i = {}, c10r = {}, c10i = {};

#if defined(HAVE_TDM) && defined(USE_WMMA_F32)
  __shared__ float sA[3 * 16 * NN];   // 12 KB: [mat][row][col]
  if (wave == 0) {
    unsigned long long ga = (unsigned long long)(const void*)(A0 + (size_t)h0 * NN);
    unsigned lds_off = (unsigned)(unsigned long long)(uintptr_t)&sA[0]; // LDS aperture: low 32 bits
    // D# group 0: count=1, lds_addr, 57-bit global_addr, type=2 ("image")
    v4u g0 = { 1u, lds_off,
               (unsigned)(ga & 0xFFFFFFFFull),
               (unsigned)((ga >> 32) & 0x1FFFFFFull) | (2u << 30) };
    // D# group 1: data_size=2 (4B); tensor 64 x 16 (x3 planes);
    // tile 64 x 16 x 3; dim0_stride=64; dim1_stride=H*N=32768
    v8i g1 = { (int)(2u << 16),
               (int)((64u & 0xFFFFu) << 16),
               (int)((64u >> 16) | ((16u & 0xFFFFu) << 16)),
               (int)((16u >> 16) | (64u << 16)),
               (int)(16u | (3u << 16)),
               (int)64,
               (int)(((32768u & 0xFFFFu) << 16)),
               (int)(32768u >> 16) };
    // D# group 2: tensor_dim2=3; tile_dim3=0 (unused). group 3: zeros.
    v4i g2 = { 3, 0, 0, 0 };
    v4i g3 = { 0, 0, 0, 0 };
#if __clang_major__ >= 23
    v8i g4 = { 0, 0, 0, 0, 0, 0, 0, 0 };
    __builtin_amdgcn_tensor_load_to_lds(g0, g1, g2, g3, g4, 0);
#else
    __builtin_amdgcn_tensor_load_to_lds(g0, g1, g2, g3, 0);
#endif
#if __has_builtin(__builtin_amdgcn_s_wait_tensorcnt)
    __builtin_amdgcn_s_wait_tensorcnt(0);
#else
    asm volatile("s_wait_tensorcnt 0x0" ::: "memory");
#endif
  }
  __syncthreads();
  const float2* sa0 = (const float2*)(sA + 0 * 1024 + lr * NN + 2 * hf);
  const float2* sa1 = (const float2*)(sA + 1 * 1024 + lr * NN + 2 * hf);
  const float2* sa2 = (const float2*)(sA + 2 * 1024 + lr * NN + 2 * hf);
  __builtin_prefetch((const void*)pbr, 0, 0);
  __builtin_prefetch((const void*)pbi, 0, 0);
#pragma unroll
  for (int kk = 0; kk < 16; ++kk) {   // K = 64 = 16 steps of 4
    float2 t;
    v2f a0v, a1v, a2v, brv, biv;
    t = sa0[kk * 2]; a0v.x = t.x; a0v.y = t.y;   // ds_load_b64 from staged A
    t = sa1[kk * 2]; a1v.x = t.x; a1v.y = t.y;
    t = sa2[kk * 2]; a2v.x = t.x; a2v.y = t.y;
    t = pbr[kk * 2]; brv.x = t.x; brv.y = t.y;
    t = pbi[kk * 2]; biv.x = t.x; biv.y = t.y;
    c00r = __builtin_amdgcn_wmma_f32_16x16x4_f32(false, a0v, false, brv, (short)0, c00r, false, false);
    c00i = __builtin_amdgcn_wmma_f32_16x16x4_f32(false, a0v, false, biv, (short)0, c00i, false, false);
    c01r = __builtin_amdgcn_wmma_f32_16x16x4_f32(false, a1v, false, brv, (short)0, c01r, false, false);
    c01i = __builtin_amdgcn_wmma_f32_16x16x4_f32(false, a1v, false, biv, (short)0, c01i, false, false);
    c10r = __builtin_amdgcn_wmma_f32_16x16x4_f32(false, a2v, false, brv, (short)0, c10r, false, false);
    c10i = __builtin_amdgcn_wmma_f32_16x16x4_f32(false, a2v, false, biv, (short)0, c10i, false, false);
  }
#elif defined(USE_WMMA_F32)
  const float2* pa0 = (const float2*)(A0 + (h0 + lr) * NN + 2 * hf);
  const float2* pa1 = (const float2*)(A1 + (h0 + lr) * NN + 2 * hf);
  const float2* pa2 = (const float2*)(A2 + (h0 + lr) * NN + 2 * hf);
  __builtin_prefetch((const void*)pbr, 0, 0);
  __builtin_prefetch((const void*)pbi, 0, 0);
#pragma unroll
  for (int kk = 0; kk < 16; ++kk) {
    float2 t;
    v2f a0v, a1v, a2v, brv, biv;
    t = pa0[kk * 2]; a0v.x = t.x; a0v.y = t.y;
    t = pa1[kk * 2]; a1v.x = t.x; a1v.y = t.y;
    t = pa2[kk * 2]; a2v.x = t.x; a2v.y = t.y;
    t = pbr[kk * 2]; brv.x = t.x; brv.y = t.y;
    t = pbi[kk * 2]; biv.x = t.x; biv.y = t.y;
    c00r = __builtin_amdgcn_wmma_f32_16x16x4_f32(false, a0v, false, brv, (short)0, c00r, false, false);
    c00i = __builtin_amdgcn_wmma_f32_16x16x4_f32(false, a0v, false, biv, (short)0, c00i, false, false);
    c01r = __builtin_amdgcn_wmma_f32_16x16x4_f32(false, a1v, false, brv, (short)0, c01r, false, false);
    c01i = __builtin_amdgcn_wmma_f32_16x16x4_f32(false, a1v, false, biv, (short)0, c01i, false, false);
    c10r = __builtin_amdgcn_wmma_f32_16x16x4_f32(false, a2v, false, brv, (short)0, c10r, false, false);
    c10i = __builtin_amdgcn_wmma_f32_16x16x4_f32(false, a2v, false, biv, (short)0, c10i, false, false);
  }
#else
  // Correct scalar fallback matching the WMMA C/D layout:
  for (int v = 0; v < 8; ++v) {
    int m = v + 8 * hf;
    const float* ra0 = A0 + (h0 + m) * NN;
    const float* ra1 = A1 + (h0 + m) * NN;
    const float* ra2 = A2 + (h0 + m) * NN;
    const float* rbr = cr + (l0 + lr) * NN;
    const float* rbi = ci + (l0 + lr) * NN;
    float s00r = 0, s00i = 0, s01r = 0, s01i = 0, s10r = 0, s10i = 0;
    for (int k = 0; k < NN; ++k) {
      float br = rbr[k], bi = rbi[k];
      s00r = fmaf(ra0[k], br, s00r); s00i = fmaf(ra0[k], bi, s00i);
      s01r = fmaf(ra1[k], br, s01r); s01i = fmaf(ra1[k], bi, s01i);
      s10r = fmaf(ra2[k], br, s10r); s10i = fmaf(ra2[k], bi, s10i);
    }
    c00r[v] = s00r; c00i[v] = s00i; c01r[v] = s01r;
    c01i[v] = s01i; c10r[v] = s10r; c10i[v] = s10i;
  }
#endif

  // Epilogue: atRoots = w2 * (k00 - k01*k10/(1+k11))
  int l = l0 + lr;
  float k11r = k11[2 * l], k11i = k11[2 * l + 1];
  float w2r = w2[2 * l], w2i = w2[2 * l + 1];
  float dr = 1.f + k11r, di = k11i;
  float dinv = 1.f / (dr * dr + di * di);
#pragma unroll
  for (int v = 0; v < 8; ++v) {
    int h = h0 + v + 8 * hf;
    float nr = c01r[v] * c10r[v] - c01i[v] * c10i[v];
    float ni = c01r[v] * c10i[v] + c01i[v] * c10r[v];
    float tr = (nr * dr + ni * di) * dinv;
    float ti = (ni * dr - nr * di) * dinv;
    float er = c00r[v] - tr, ei = c00i[v] - ti;
    float ar = w2r * er - w2i * ei;
    float ai = w2r * ei + w2i * er;
    outAt[(size_t)h * L2 + l] = make_float2(ar, ai);
  }
}

// ---------------------------------------------------------------------------
// Stockham radix-4 pass over global memory (192MB L2 holds the 128MB set).
// lQ = log2(Nfft/4); lNs = log2(current span). Row stride fixed at 2L.
// ---------------------------------------------------------------------------
__global__ __launch_bounds__(256) void fft_pass_r4(
    const float2* __restrict__ in, float2* __restrict__ out,
    int lQ, int lNs, float dir) {
  int gid = blockIdx.x * 256 + threadIdx.x;
  int quarter = 1 << lQ;
  int j = gid & (quarter - 1);
  int h = gid >> lQ;
  const float2* src = in + (size_t)h * L2;
  float2* dst = out + (size_t)h * L2;

  int Ns = 1 << lNs;
  int k = j & (Ns - 1);
  int grp = j >> lNs;
  int db = (grp << (lNs + 2)) + k;

  float2 v0 = src[j];
  float2 v1 = src[j + quarter];
  float2 v2 = src[j + 2 * quarter];
  float2 v3 = src[j + 3 * quarter];

  float ang = dir * 6.28318530717958647692f * (float)k / (float)(Ns << 2);
  float s1, c1, s2, c2, s3, c3;
  __sincosf(ang, &s1, &c1);
  __sincosf(2.f * ang, &s2, &c2);
  __sincosf(3.f * ang, &s3, &c3);
  v1 = cmulf(v1, make_float2(c1, s1));
  v2 = cmulf(v2, make_float2(c2, s2));
  v3 = cmulf(v3, make_float2(c3, s3));

  float2 t0 = caddf(v0, v2), t1 = csubf(v0, v2);
  float2 t2 = caddf(v1, v3), t3 = csubf(v1, v3);
  float2 w = (dir < 0.f) ? make_float2(t3.y, -t3.x) : make_float2(-t3.y, t3.x);
  dst[db] = caddf(t0, t2);
  dst[db + Ns] = caddf(t1, w);
  dst[db + 2 * Ns] = csubf(t0, t2);
  dst[db + 3 * Ns] = csubf(t1, w);
}

// Final radix-2 pass for Nfft = 32768 (= 4^7 * 2)
__global__ __launch_bounds__(256) void fft_pass_r2(
    const float2* __restrict__ in, float2* __restrict__ out,
    int lH, int lNs, float dir) {
  int gid = blockIdx.x * 256 + threadIdx.x;
  int halfn = 1 << lH;
  int j = gid & (halfn - 1);
  int h = gid >> lH;
  const float2* src = in + (size_t)h * L2;
  float2* dst = out + (size_t)h * L2;

  int Ns = 1 << lNs;
  int k = j & (Ns - 1);
  int grp = j >> lNs;
  int db = (grp << (lNs + 1)) + k;

  float ang = dir * 6.28318530717958647692f * (float)k / (float)(Ns << 1);
  float s, c;
  __sincosf(ang, &s, &c);
  float2 v0 = src[j];
  float2 v1 = cmulf(src[j + halfn], make_float2(c, s));
  dst[db] = caddf(v0, v1);
  dst[db + Ns] = csubf(v0, v1);
}

// ---------------------------------------------------------------------------
// Elementwise helpers
// ---------------------------------------------------------------------------
__global__ __launch_bounds__(256) void s4_pad_scale(
    const float2* __restrict__ src, float2* __restrict__ dst, float scale) {
  int idx = blockIdx.x * 256 + threadIdx.x;  // H*2L threads
  int h = idx >> 15, t = idx & (L2 - 1);
  float2 v = make_float2(0.f, 0.f);
  if (t < LL) {
    float2 s = src[(size_t)h * L2 + t];
    v = make_float2(s.x * scale, s.y * scale);
  }
  dst[(size_t)h * L2 + t] = v;
}

__global__ __launch_bounds__(256) void s4_load_y(
    const float* __restrict__ y, float2* __restrict__ dst) {
  int idx = blockIdx.x * 256 + threadIdx.x;  // H*2L threads
  int h = idx >> 15, t = idx & (L2 - 1);
  float re = (t < LL) ? y[(size_t)h * LL + t] : 0.f;
  dst[(size_t)h * L2 + t] = make_float2(re, 0.f);
}

__global__ __launch_bounds__(256) void s4_pointwise_mul(
    float2* __restrict__ a, const float2* __restrict__ b, float scale) {
  size_t idx = (size_t)blockIdx.x * 256 + threadIdx.x;  // H*2L threads
  float2 av = a[idx], bv = b[idx];
  float2 r = cmulf(av, bv);
  a[idx] = make_float2(r.x * scale, r.y * scale);
}

__global__ __launch_bounds__(256) void s4_writeout(
    const float2* __restrict__ src, float* __restrict__ out) {
  int idx = blockIdx.x * 256 + threadIdx.x;  // H*L threads
  int h = idx >> 14, l = idx & (LL - 1);
  float2 v = src[(size_t)h * L2 + l];
  out[2 * (size_t)idx] = v.x;
  out[2 * (size_t)idx + 1] = v.y;
}

// ---------------------------------------------------------------------------
extern "C" void kernel_launch(void* const* d_in, const int* in_sizes, int n_in,
                              void* d_out, int out_size, void* d_ws, size_t ws_size,
                              hipStream_t stream) {
  (void)in_sizes; (void)n_in; (void)out_size; (void)ws_size;
  const float* y  = (const float*)d_in[0];
  const float* Bm = (const float*)d_in[1];
  const float* Cm = (const float*)d_in[2];
  const float* G  = (const float*)d_in[3];  // complex64 interleaved
  const float* p  = (const float*)d_in[4];
  const float* q  = (const float*)d_in[5];
  const float* Mm = (const float*)d_in[6];
  float* out = (float*)d_out;

  float* ws = (float*)d_ws;
  size_t o = 0;
  float* A0 = ws + o; o += (size_t)HH * NN;   // A0/A1/A2 contiguous: 3-D TDM tensor
  float* A1 = ws + o; o += (size_t)HH * NN;
  float* A2 = ws + o; o += (size_t)HH * NN;
  float* qp = ws + o; o += NN;
  float* k11 = ws + o; o += 2 * (size_t)LL;
  float* w2  = ws + o; o += 2 * (size_t)LL;
  float* cr  = ws + o; o += (size_t)LL * NN;
  float* ci  = ws + o; o += (size_t)LL * NN;
  o = (o + 3) & ~(size_t)3;
  float2* bufA = (float2*)(ws + o); o += (size_t)HH * L2 * 2;
  float2* bufB = (float2*)(ws + o); o += (size_t)HH * L2 * 2;
  float2* bufC = (float2*)(ws + o);

  // Stage 1: coefficients
  s4_prep<<<(HH * NN) / 256, 256, 0, stream>>>(Bm, Cm, Mm, p, q, A0, A1, A2, qp);
  s4_cauchy_init<<<LL, 64, 0, stream>>>(G, qp, cr, ci, k11, w2);

  // Stage 2: WMMA GEMM + epilogue -> atRoots in bufA (first L of each 2L row)
  s4_cauchy_gemm<<<4096, 256, 0, stream>>>(A0, A1, A2, cr, ci, k11, w2, bufA);

  // Stage 3: K = fft_L(atRoots)/L  (7 radix-4 passes, ping-pong A<->B)
  float2 *pA = bufA, *pB = bufB;
  int lNs = 0;
  for (int i = 0; i < 7; ++i) {
    fft_pass_r4<<<(HH * (LL / 4)) / 256, 256, 0, stream>>>(pA, pB, 12, lNs, -1.f);
    float2* t = pA; pA = pB; pB = t; lNs += 2;
  }
  s4_pad_scale<<<(HH * L2) / 256, 256, 0, stream>>>(pA, bufA, 1.f / (float)LL);

  // Stage 4: Kf2 = fft_2L(K): 7 r4 + 1 r2, ends back in bufA
  pA = bufA; pB = bufB; lNs = 0;
  for (int i = 0; i < 7; ++i) {
    fft_pass_r4<<<(HH * (L2 / 4)) / 256, 256, 0, stream>>>(pA, pB, 13, lNs, -1.f);
    float2* t = pA; pA = pB; pB = t; lNs += 2;
  }
  fft_pass_r2<<<(HH * (L2 / 2)) / 256, 256, 0, stream>>>(pA, pB, 14, 14, -1.f);
  { float2* t = pA; pA = pB; pB = t; }  // Kf2 == bufA

  // Stage 5: Yf = fft_2L(pad(y)), ping-pong B<->C, ends in bufB
  s4_load_y<<<(HH * L2) / 256, 256, 0, stream>>>(y, bufB);
  pA = bufB; pB = bufC; lNs = 0;
  for (int i = 0; i < 7; ++i) {
    fft_pass_r4<<<(HH * (L2 / 4)) / 256, 256, 0, stream>>>(pA, pB, 13, lNs, -1.f);
    float2* t = pA; pA = pB; pB = t; lNs += 2;
  }
  fft_pass_r2<<<(HH * (L2 / 2)) / 256, 256, 0, stream>>>(pA, pB, 14, 14, -1.f);
  { float2* t = pA; pA = pB; pB = t; }  // Yf == bufB

  // Stage 6: P = Yf * Kf2 * (1/2L)  (folds ifft scaling)
  s4_pointwise_mul<<<(HH * L2) / 256, 256, 0, stream>>>(bufB, bufA, 1.f / (float)L2);

  // Stage 7: conv = ifft_2L(P), ping-pong B<->C, ends in bufB
  pA = bufB; pB = bufC; lNs = 0;
  for (int i = 0; i < 7; ++i) {
    fft_pass_r4<<<(HH * (L2 / 4)) / 256, 256, 0, stream>>>(pA, pB, 13, lNs, +1.f);
    float2* t = pA; pA = pB; pB = t; lNs += 2;
  }
  fft_pass_r2<<<(HH * (L2 / 2)) / 256, 256, 0, stream>>>(pA, pB, 14, 14, +1.f);
  { float2* t = pA; pA = pB; pB = t; }

  // Stage 8: write first L columns, interleaved complex
  s4_writeout<<<(HH * LL) / 256, 256, 0, stream>>>(bufB, out);
}